// NW3_73572789780680
// MI455X (gfx1250) — compile-verified
//
#include <hip/hip_runtime.h>
#include <hip/hip_bf16.h>

typedef float v2f __attribute__((ext_vector_type(2)));
typedef float v8f __attribute__((ext_vector_type(8)));
typedef int   v4i __attribute__((ext_vector_type(4)));

// gfx1250 async global->LDS copy (ASYNCcnt-tracked), if the toolchain declares it.
#if defined(__has_builtin)
#if __has_builtin(__builtin_amdgcn_global_load_async_to_lds_b128) && \
    __has_builtin(__builtin_amdgcn_s_wait_asynccnt)
#define USE_ASYNC_LDS 1
#endif
#endif
#ifndef USE_ASYNC_LDS
#define USE_ASYNC_LDS 0
#endif

typedef __attribute__((address_space(1))) v4i g_v4i;  // global int4
typedef __attribute__((address_space(3))) v4i l_v4i;  // LDS int4

static __device__ __forceinline__ void stage_b128(const float* gsrc, float* ldst) {
#if USE_ASYNC_LDS
  __builtin_amdgcn_global_load_async_to_lds_b128((g_v4i*)gsrc, (l_v4i*)ldst, 0, 0);
#else
  *(float4*)ldst = *(const float4*)gsrc;
#endif
}
static __device__ __forceinline__ void stage_wait() {
#if USE_ASYNC_LDS
  __builtin_amdgcn_s_wait_asynccnt(0);
#endif
}

static __device__ __forceinline__ v8f wmma_f32_4(v2f a, v2f b, v8f c) {
  // D = A(16x4 f32) * B(4x16 f32) + C(16x16 f32)
  return __builtin_amdgcn_wmma_f32_16x16x4_f32(false, a, false, b, (short)0, c,
                                               false, false);
}

constexpr int NB  = 8;
constexpr int NKk = 2048;
constexpr int NQq = 2048;
constexpr int DK  = 3;
constexpr int DV  = 8;
constexpr int H   = 128;

// -------------------------------------------------------------------------
// 3-layer MLP head: out = relu(relu(x@w1+b1)@w2+b2)@w3+b3, optionally * x.
// One wave handles a 16-token tile; middle 128x128 layer uses f32 WMMA.
// -------------------------------------------------------------------------
template <int FIO, bool SCALE>
__global__ __launch_bounds__(128) void mlp_head_kernel(
    const float* __restrict__ x,
    const float* __restrict__ w1, const float* __restrict__ b1,
    const float* __restrict__ w2, const float* __restrict__ b2,
    const float* __restrict__ w3, const float* __restrict__ b3,
    float* __restrict__ out) {
  constexpr int WAVES = 4;
  constexpr int ROW = 132;  // padded row stride (floats) -> conflict-free A reads
  __shared__ __align__(16) float h_lds[WAVES][16 * ROW];
  __shared__ __align__(16) float x_lds[WAVES][16 * FIO];

  const int lane = threadIdx.x & 31;
  const int wave = threadIdx.x >> 5;
  const int tok0 = (blockIdx.x * WAVES + wave) * 16;
  float* hl = h_lds[wave];
  float* xl = x_lds[wave];

  // stage this wave's 16xFIO input tile (contiguous, 16B-aligned): 4*FIO float4s
  if (lane < 4 * FIO)
    stage_b128(x + (size_t)tok0 * FIO + lane * 4, xl + lane * 4);
  stage_wait();
  __syncthreads();

  // ---- layer 1: FIO -> 128, VALU ----
#pragma unroll
  for (int jb = 0; jb < 4; ++jb) {
    const int j = jb * 32 + lane;
    float wc[FIO];
#pragma unroll
    for (int d = 0; d < FIO; ++d) wc[d] = w1[d * H + j];
    const float bj = b1[j];
#pragma unroll
    for (int m = 0; m < 16; ++m) {
      float s = bj;
#pragma unroll
      for (int d = 0; d < FIO; ++d) s = fmaf(xl[m * FIO + d], wc[d], s);
      hl[m * ROW + j] = fmaxf(s, 0.0f);
    }
  }
  __syncthreads();

  // ---- layer 2: 128 -> 128 via V_WMMA_F32_16X16X4_F32 ----
  // A frag (16x4): lane l -> M = l&15, K = 2*(l>=16) + v
  // B frag (4x16): lane l -> K = 2*(l>=16) + v, N = l&15
  // C/D:           lane l, vgpr r -> M = 8*(l>=16) + r, N = l&15
  const int m16 = lane & 15;
  const int khi = (lane >> 4) << 1;
  v8f acc[8];
#pragma unroll
  for (int n = 0; n < 8; ++n) {
    const float bv = b2[n * 16 + m16];  // bias depends on output column only
#pragma unroll
    for (int r = 0; r < 8; ++r) acc[n][r] = bv;
  }
  for (int k = 0; k < 32; ++k) {
    v2f a;
    a.x = hl[m16 * ROW + k * 4 + khi];
    a.y = hl[m16 * ROW + k * 4 + khi + 1];
    const float* wp = w2 + (size_t)(k * 4 + khi) * H + m16;
#pragma unroll
    for (int n = 0; n < 8; ++n) {
      v2f bf;
      bf.x = wp[n * 16];
      bf.y = wp[n * 16 + H];
      acc[n] = wmma_f32_4(a, bf, acc[n]);
    }
  }
  // relu + write h2 back over h1 (same wave; LDS ops are in-order per wave)
  const int mbase = (lane >> 4) << 3;
#pragma unroll
  for (int n = 0; n < 8; ++n)
#pragma unroll
    for (int r = 0; r < 8; ++r)
      hl[(mbase + r) * ROW + n * 16 + m16] = fmaxf(acc[n][r], 0.0f);
  __syncthreads();

  // ---- layer 3: 128 -> FIO, VALU; fuse optional elementwise scale ----
  {
    const int m = lane & 15;
    const int d0 = lane >> 4;
    for (int d = d0; d < FIO; d += 2) {
      float s = b3[d];
#pragma unroll 8
      for (int j = 0; j < H; ++j) s = fmaf(hl[m * ROW + j], w3[j * FIO + d], s);
      if (SCALE) s *= xl[m * FIO + d];
      out[(size_t)(tok0 + m) * FIO + d] = s;
    }
  }
}

// -------------------------------------------------------------------------
// Distance attention: logits = -0.5*(L1dist)^2, softmax over keys, ctx=attn^T@V
// One wave per 16 queries. Per 4-key step one f32 WMMA accumulates both
// ctx (cols 0..7) and the softmax denominator (ones-column at N=8).
// -------------------------------------------------------------------------
__global__ __launch_bounds__(256) void attn_ctx_kernel(
    const float* __restrict__ ksc,   // [B,NK,3] scaled keys
    const float* __restrict__ qsc,   // [B,NQ,3] scaled queries
    const float* __restrict__ val,   // [B,NK,8]
    float* __restrict__ ctx) {       // [B,NQ,8]
  constexpr int TK = 128;
  __shared__ __align__(16) float k_lds[TK * DK];  // 384 floats = 96 x b128
  __shared__ __align__(16) float v_lds[TK * DV];  // 1024 floats = 256 x b128

  const int lane = threadIdx.x & 31;
  const int wave = threadIdx.x >> 5;
  constexpr int QPB = 8 * 16;            // 128 queries per block (8 waves)
  constexpr int BLKPER = NQq / QPB;      // 16 blocks per batch
  const int b = blockIdx.x / BLKPER;
  const int q0 = (blockIdx.x % BLKPER) * QPB + wave * 16;
  const int m16 = lane & 15;
  const int khi = (lane >> 4) << 1;

  const float* qp = qsc + ((size_t)b * NQq + q0 + m16) * DK;
  const float qc0 = qp[0], qc1 = qp[1], qc2 = qp[2];

  // loop-invariant B-fragment select: lanes 0..7 take V, lane 8 takes 1.0,
  // lanes 9..15 take 0.0; index clamped so the LDS load is unconditional.
  const bool isv = (m16 < DV);
  const float cpad = (m16 == DV) ? 1.0f : 0.0f;
  const int coff = m16 & (DV - 1);

  v8f acc = {};  // C[m][0..7]=ctx, C[m][8]=sum of weights

  for (int kt = 0; kt < NKk; kt += TK) {
    __syncthreads();
    // stage tiles as b128: V is one float4/thread, K is one float4 for tid<96
    stage_b128(val + ((size_t)b * NKk + kt) * DV + threadIdx.x * 4,
               v_lds + threadIdx.x * 4);
    if (threadIdx.x < (TK * DK) / 4)
      stage_b128(ksc + ((size_t)b * NKk + kt) * DK + threadIdx.x * 4,
                 k_lds + threadIdx.x * 4);
    stage_wait();
    __syncthreads();

    for (int s = 0; s < TK; s += 4) {
      const int ka = s + khi;
      const int kb = ka + 1;
      float da = fabsf(k_lds[ka * DK + 0] - qc0) +
                 fabsf(k_lds[ka * DK + 1] - qc1) +
                 fabsf(k_lds[ka * DK + 2] - qc2);
      float db = fabsf(k_lds[kb * DK + 0] - qc0) +
                 fabsf(k_lds[kb * DK + 1] - qc1) +
                 fabsf(k_lds[kb * DK + 2] - qc2);
      v2f a;
      a.x = __expf(-0.5f * da * da);   // logits <= 0: exp cannot overflow
      a.y = __expf(-0.5f * db * db);
      // unconditional in-bounds LDS loads + v_cndmask select (no exec masking)
      const float lva = v_lds[ka * DV + coff];
      const float lvb = v_lds[kb * DV + coff];
      v2f bf;  // B(4x16): V columns 0..7, ones-column at 8, zeros 9..15
      bf.x = isv ? lva : cpad;
      bf.y = isv ? lvb : cpad;
      acc = wmma_f32_4(a, bf, acc);
    }
  }

  const int mbase = (lane >> 4) << 3;
#pragma unroll
  for (int r = 0; r < 8; ++r) {
    const float ssum = __shfl(acc[r], (lane & 16) | 8, 32);  // C[m][8]
    if (m16 < DV) {
      const float rinv = 1.0f / fmaxf(ssum, 1e-35f);
      ctx[((size_t)b * NQq + q0 + mbase + r) * DV + m16] = acc[r] * rinv;
    }
  }
}

// -------------------------------------------------------------------------
extern "C" void kernel_launch(void* const* d_in, const int* in_sizes, int n_in,
                              void* d_out, int out_size, void* d_ws,
                              size_t ws_size, hipStream_t stream) {
  const float* KEY   = (const float*)d_in[0];
  const float* VALUE = (const float*)d_in[1];
  const float* QUERY = (const float*)d_in[2];
  const float* wk1 = (const float*)d_in[3];  const float* bk1 = (const float*)d_in[4];
  const float* wk2 = (const float*)d_in[5];  const float* bk2 = (const float*)d_in[6];
  const float* wk3 = (const float*)d_in[7];  const float* bk3 = (const float*)d_in[8];
  const float* wq1 = (const float*)d_in[9];  const float* bq1 = (const float*)d_in[10];
  const float* wq2 = (const float*)d_in[11]; const float* bq2 = (const float*)d_in[12];
  const float* wq3 = (const float*)d_in[13]; const float* bq3 = (const float*)d_in[14];
  const float* wo1 = (const float*)d_in[15]; const float* bo1 = (const float*)d_in[16];
  const float* wo2 = (const float*)d_in[17]; const float* bo2 = (const float*)d_in[18];
  const float* wo3 = (const float*)d_in[19]; const float* bo3 = (const float*)d_in[20];

  float* ws  = (float*)d_ws;
  float* ksc = ws;                                   // [B,NK,3]
  float* qsc = ksc + (size_t)NB * NKk * DK;          // [B,NQ,3]
  float* ctx = qsc + (size_t)NB * NQq * DK;          // [B,NQ,8]

  const int mlp_blocks = (NB * NKk) / 64;  // 64 tokens per 128-thread block
  mlp_head_kernel<DK, true><<<mlp_blocks, 128, 0, stream>>>(
      KEY, wk1, bk1, wk2, bk2, wk3, bk3, ksc);
  mlp_head_kernel<DK, true><<<mlp_blocks, 128, 0, stream>>>(
      QUERY, wq1, bq1, wq2, bq2, wq3, bq3, qsc);
  attn_ctx_kernel<<<NB * (NQq / 128), 256, 0, stream>>>(ksc, qsc, VALUE, ctx);
  mlp_head_kernel<DV, false><<<(NB * NQq) / 64, 128, 0, stream>>>(
      ctx, wo1, bo1, wo2, bo2, wo3, bo3, (float*)d_out);
}